// StackedLinear_74801150427257
// MI455X (gfx1250) — compile-verified
//
#include <hip/hip_runtime.h>

#define NS   8
#define IN   512
#define OUT  32

typedef __attribute__((ext_vector_type(2))) float v2f;
typedef __attribute__((ext_vector_type(8))) float v8f;

// ---- workspace layout (ints): [0..7]=cnt  [8..15]=cursor  [16..23]=off  [32..]=rows (B ints) ----

__global__ void sl_init_kernel(int* ws) {
    if (threadIdx.x < 16) ws[threadIdx.x] = 0;   // cnt + cursor
}

__global__ void sl_count_kernel(const int* __restrict__ idx, int* __restrict__ cnt, int n) {
    int b = blockIdx.x * blockDim.x + threadIdx.x;
    if (b < n) atomicAdd(&cnt[idx[b]], 1);
}

__global__ void sl_scan_kernel(const int* __restrict__ cnt, int* __restrict__ off) {
    if (threadIdx.x == 0) {
        int a = 0;
        for (int i = 0; i < NS; ++i) { off[i] = a; a += cnt[i]; }
    }
}

__global__ void sl_place_kernel(const int* __restrict__ idx, int* __restrict__ cursor,
                                const int* __restrict__ off, int* __restrict__ rows, int n) {
    int b = blockIdx.x * blockDim.x + threadIdx.x;
    if (b < n) {
        int s = idx[b];
        int p = atomicAdd(&cursor[s], 1);
        rows[off[s] + p] = b;
    }
}

// One workgroup = 8 waves = one stack, 128 gathered rows.
// wave w: 16x32 output tile via two v_wmma_f32_16x16x4_f32 accumulators.
__global__ __launch_bounds__(256)
void sl_gemm_kernel(const float* __restrict__ x, const float* __restrict__ w,
                    const float* __restrict__ bias,
                    const int* __restrict__ cnt, const int* __restrict__ off,
                    const int* __restrict__ rows, float* __restrict__ out) {
    __shared__ float wlds[OUT * IN];   // 64 KB: this stack's full weight

    // Map flat block id -> (stack s, local 128-row tile)
    int wg = blockIdx.x;
    int s = -1, local = 0, acc_t = 0;
    for (int i = 0; i < NS; ++i) {
        int t = (cnt[i] + 127) >> 7;
        if (s < 0 && wg < acc_t + t) { s = i; local = wg - acc_t; }
        acc_t += t;
    }
    if (s < 0) return;                      // uniform per block: safe before barrier
    const int c       = cnt[s];
    const int segBase = off[s];

    // Cooperatively stage weight[s] (32x512 f32, row-major) into LDS
    const float4* wsrc = (const float4*)(w + (long)s * OUT * IN);
    float4* wdst = (float4*)wlds;
    for (int i = threadIdx.x; i < (OUT * IN) / 4; i += 256) wdst[i] = wsrc[i];
    __syncthreads();

    const int lane  = threadIdx.x & 31;
    const int wave  = threadIdx.x >> 5;
    const int m     = lane & 15;            // N column / A row index within tile
    const int half  = lane >> 4;            // lane half selects K{0,1} vs K{2,3}
    const int tilePos = local * 128 + wave * 16;

    // Gather this lane's A row (clamped for partial tiles; stores are predicated)
    int rpos = tilePos + m;
    int rcl  = rpos < c ? rpos : c - 1;
    int rowIdx = rows[segBase + rcl];
    const float* arow = x + (long)rowIdx * IN;

    const float* b0p = &wlds[m * IN];          // B fragment, N-tile 0 (cols 0..15)
    const float* b1p = &wlds[(16 + m) * IN];   // B fragment, N-tile 1 (cols 16..31)
    const int koff = 2 * half;                 // A/B 16x4 f32 layout: v0,v1 = K(koff), K(koff+1)

    v8f acc0 = {};
    v8f acc1 = {};
    #pragma unroll 4
    for (int k0 = 0; k0 < IN; k0 += 4) {
        v2f a  = *(const v2f*)(arow + k0 + koff);
        v2f b0 = *(const v2f*)(b0p + k0 + koff);
        v2f b1 = *(const v2f*)(b1p + k0 + koff);
        acc0 = __builtin_amdgcn_wmma_f32_16x16x4_f32(false, a, false, b0,
                                                     (short)0, acc0, false, false);
        acc1 = __builtin_amdgcn_wmma_f32_16x16x4_f32(false, a, false, b1,
                                                     (short)0, acc1, false, false);
    }

    // C/D layout: acc[r] holds row (r + 8*half), col m of the 16x16 tile
    float bv0 = bias[s * OUT + m];
    float bv1 = bias[s * OUT + 16 + m];
    #pragma unroll
    for (int r = 0; r < 8; ++r) {
        int pos = tilePos + r + 8 * half;
        if (pos < c) {
            int ob = rows[segBase + pos];
            out[(long)ob * OUT + m]      = acc0[r] + bv0;
            out[(long)ob * OUT + 16 + m] = acc1[r] + bv1;
        }
    }
}

extern "C" void kernel_launch(void* const* d_in, const int* in_sizes, int n_in,
                              void* d_out, int out_size, void* d_ws, size_t ws_size,
                              hipStream_t stream) {
    const float* x    = (const float*)d_in[0];
    const int*   idx  = (const int*)d_in[1];
    const float* w    = (const float*)d_in[2];
    const float* bias = (const float*)d_in[3];
    float* out = (float*)d_out;
    const int n = in_sizes[1];               // batch size B

    int* ws     = (int*)d_ws;
    int* cnt    = ws;
    int* cursor = ws + 8;
    int* off    = ws + 16;
    int* rows   = ws + 32;

    const int nb = (n + 255) / 256;
    sl_init_kernel<<<1, 32, 0, stream>>>(ws);
    sl_count_kernel<<<nb, 256, 0, stream>>>(idx, cnt, n);
    sl_scan_kernel<<<1, 32, 0, stream>>>(cnt, off);
    sl_place_kernel<<<nb, 256, 0, stream>>>(idx, cursor, off, rows, n);

    const int maxTiles = (n + 127) / 128 + NS;   // upper bound on sum of ceil(cnt_s/128)
    sl_gemm_kernel<<<maxTiles, 256, 0, stream>>>(x, w, bias, cnt, off, rows, out);
}